// StudentAttentionModel_72713796321750
// MI455X (gfx1250) — compile-verified
//
#include <hip/hip_runtime.h>
#include <cstdint>

// Problem shape (fixed by the reference): B=32768, L=2, D=1024, DK=1.
namespace {
constexpr int kB            = 32768;
constexpr int kD            = 1024;
constexpr int kWavesPerBlk  = 8;                    // wave32 -> 256 threads/block
constexpr int kBlockThreads = kWavesPerBlk * 32;
constexpr int kV4PerLane    = kD / 4 / 32;          // 8 vec4 per lane per row
constexpr float kInvSqrtD   = 0.03125f;             // 1/sqrt(1024) = 1/32
}

// Native clang vector so __builtin_nontemporal_* accepts it (HIP float4 is a struct).
typedef float v4f __attribute__((ext_vector_type(4)));

__global__ __launch_bounds__(kBlockThreads, 1) void
StudentAttention_b2_stream_kernel(const float* __restrict__ X,
                                  const float* __restrict__ Wq,
                                  float* __restrict__ Y) {
    __shared__ float s_wq[kD];  // raw W_Q staged via async DMA (4 KB)

    const int tid = threadIdx.x;

    // ---- Stage W_Q into LDS with CDNA5 async global->LDS DMA (one b128/thread) ----
    {
        const uint64_t gaddr = (uint64_t)(uintptr_t)(Wq + tid * 4);
        // Low 32 bits of a flat LDS pointer are the wave-relative LDS byte address
        // (ISA 10.2); hardware adds LDS_BASE itself.
        const uint32_t ldsa = (uint32_t)(uintptr_t)(&s_wq[tid * 4]);
        asm volatile("global_load_async_to_lds_b128 %0, %1, off"
                     :: "v"(ldsa), "v"(gaddr)
                     : "memory");
#if __has_builtin(__builtin_amdgcn_s_wait_asynccnt)
        __builtin_amdgcn_s_wait_asynccnt(0);
#else
        asm volatile("s_wait_asynccnt 0x0" ::: "memory");
#endif
    }
    __syncthreads();

    // ---- One wave32 per batch ----
    const int wave = tid >> 5;
    const int lane = tid & 31;
    const int b    = blockIdx.x * kWavesPerBlk + wave;   // grid sized so b < kB

    const v4f* __restrict__ xr0 = (const v4f*)(X + (size_t)b * 2 * kD);
    const v4f* __restrict__ xr1 = xr0 + kD / 4;
    const v4f* __restrict__ wv  = (const v4f*)s_wq;

    v4f p[kV4PerLane];   // X_pos row 0 (X + 1/32), kept in registers
    v4f m[kV4PerLane];   // X_pos row 1 (X - 1/32)
    float acc0 = 0.0f, acc1 = 0.0f;
    const float c = kInvSqrtD;

#pragma unroll
    for (int i = 0; i < kV4PerLane; ++i) {
        const int idx = lane + i * 32;                 // 512B coalesced per wave op
        v4f a  = __builtin_nontemporal_load(&xr0[idx]);
        v4f bd = __builtin_nontemporal_load(&xr1[idx]);
        const v4f w = wv[idx];
        a  += c;                                       // X_pos row 0
        bd -= c;                                       // X_pos row 1
        acc0 += a.x  * w.x + a.y  * w.y + a.z  * w.z + a.w  * w.w;
        acc1 += bd.x * w.x + bd.y * w.y + bd.z * w.z + bd.w * w.w;
        p[i] = a;
        m[i] = bd;
    }

    // Wave32 butterfly reduction: every lane ends up with the full dot products.
#pragma unroll
    for (int off = 16; off > 0; off >>= 1) {
        acc0 += __shfl_xor(acc0, off, 32);
        acc1 += __shfl_xor(acc1, off, 32);
    }

    // q = dot(X_pos, W_Q) * (1/32)  (fold the W_Q scale in once, post-reduction)
    const float q0   = acc0 * kInvSqrtD;
    const float q1   = acc1 * kInvSqrtD;
    const float diff = q0 - q1;
    // Exact 2-way softmax: row0 = softmax([q0*q0, q0*q1]) -> sigmoid(q0*diff)
    const float a00 = 1.0f / (1.0f + __expf(-(q0 * diff)));
    const float a10 = 1.0f / (1.0f + __expf(-(q1 * diff)));
    const float a01 = 1.0f - a00;
    const float a11 = 1.0f - a10;

    v4f* __restrict__ yr0 = (v4f*)(Y + (size_t)b * 2 * kD);
    v4f* __restrict__ yr1 = yr0 + kD / 4;

#pragma unroll
    for (int i = 0; i < kV4PerLane; ++i) {
        const int idx = lane + i * 32;
        const v4f a  = p[i];
        const v4f bd = m[i];
        const v4f o0 = a00 * a + a01 * bd;
        const v4f o1 = a10 * a + a11 * bd;
        __builtin_nontemporal_store(o0, &yr0[idx]);   // single-use output: NT store
        __builtin_nontemporal_store(o1, &yr1[idx]);
    }
}

extern "C" void kernel_launch(void* const* d_in, const int* in_sizes, int n_in,
                              void* d_out, int out_size, void* d_ws, size_t ws_size,
                              hipStream_t stream) {
    (void)in_sizes; (void)n_in; (void)d_ws; (void)ws_size; (void)out_size;
    const float* X  = (const float*)d_in[0];   // (32768, 2, 1024) f32
    const float* Wq = (const float*)d_in[1];   // (1024, 1) f32
    float* Y        = (float*)d_out;           // (32768, 2, 1024) f32

    const int grid = kB / kWavesPerBlk;        // 4096 blocks x 256 threads
    StudentAttention_b2_stream_kernel<<<grid, kBlockThreads, 0, stream>>>(X, Wq, Y);
}